// ConvLayer_88210038326192
// MI455X (gfx1250) — compile-verified
//
#include <hip/hip_runtime.h>

// CDNA5 / gfx1250 implicit-GEMM conv 3x3 using V_WMMA_F32_16X16X4_F32
// plus GLOBAL_LOAD_ASYNC_TO_LDS_B128 for the dense weight tile.
// GEMM view: Out[128 x 100352] = W[128 x 576] * Im2col(X)[576 x 100352] + offset.
// Block tile: M=128 (all channels) x N=64 positions; 8 waves as 4(M) x 2(N);
// each wave computes a 2x2 grid of 16x16 accumulators (32 ch x 32 pos).

typedef __attribute__((ext_vector_type(2))) float v2f;
typedef __attribute__((ext_vector_type(8))) float v8f;

namespace {
constexpr int kCin  = 64;
constexpr int kHW   = 56;
constexpr int kK    = 128;           // output channels
constexpr int kSP   = kHW * kHW;     // 3136 spatial per image
constexpr int kRdim = kCin * 9;      // 576 reduction length
constexpr int kNtot = 32 * kSP;      // 100352 output positions
constexpr int kKC   = 32;            // K-chunk staged in LDS per iteration
constexpr int kNP   = 64;            // output positions per block
constexpr int kStr  = 36;            // padded LDS row stride (dwords): 144B rows
}  // namespace

__device__ __forceinline__ void async_wtile_b128(unsigned lds_byte_off,
                                                 const float* gaddr) {
  // copies 16 bytes from global address gaddr to LDS offset lds_byte_off, tracked by ASYNCcnt.
  asm volatile("global_load_async_to_lds_b128 %0, %1, off"
               :: "v"(lds_byte_off), "v"(gaddr)
               : "memory");
}

__device__ __forceinline__ void wait_asynccnt0() {
  asm volatile("s_wait_asynccnt 0" ::: "memory");
}

__global__ __launch_bounds__(256)
void conv3x3_wmma_f32(const float* __restrict__ X, const float* __restrict__ W,
                      const float* __restrict__ Off, float* __restrict__ Out) {
  __shared__ __align__(16) float lds_w[kK * kStr];    // [128][36] weight tile
  __shared__ __align__(16) float lds_xt[kNP * kStr];  // [64][36] im2col, [p][r]

  const int tid  = threadIdx.x;
  const int lane = tid & 31;
  const int wave = tid >> 5;
  const int hf   = lane >> 4;          // half-wave select
  const int idx  = lane & 15;
  const int m00  = (wave & 3) * 32;    // wave's channel base   (4 M-groups)
  const int p00  = (wave >> 2) * 32;   // wave's position base  (2 N-groups)

  const int pbase = blockIdx.x * kNP;

  // Per-thread im2col staging coordinates (position fixed per thread).
  const int lp  = tid & 63;            // position column 0..63
  const int lr0 = tid >> 6;            // row base 0..3 (+4*i)
  const int ln  = pbase + lp;
  const int lb  = ln / kSP;
  const int lsp = ln - lb * kSP;
  const int loy = lsp / kHW;
  const int lox = lsp - loy * kHW;

  // Weight async-staging coordinates: 4 x b128 per thread per chunk.
  // float4 id q = tid + i*256 -> row = q/8 (0..127), c4 = q%8 (0..7).
  v8f c00 = {}, c01 = {}, c10 = {}, c11 = {};

  for (int chunk = 0; chunk < kRdim / kKC; ++chunk) {
    const int rbase = chunk * kKC;

    // ---- gather im2col x tile into registers (issue loads early) ----
    float xreg[8];
#pragma unroll
    for (int i = 0; i < 8; ++i) {
      const int rr = lr0 + i * 4;      // 0..31
      const int r  = rbase + rr;
      const int ci = r / 9;
      const int t9 = r - ci * 9;
      const int ky = t9 / 3;
      const int kx = t9 - ky * 3;
      const int iy = loy + ky - 1;
      const int ix = lox + kx - 1;
      float v = 0.0f;
      if ((unsigned)iy < (unsigned)kHW && (unsigned)ix < (unsigned)kHW)
        v = X[((lb * kCin + ci) * kHW + iy) * kHW + ix];
      xreg[i] = v;
    }

    __syncthreads();  // all waves done reading previous chunk's LDS

    // ---- async-stage the 128x32 weight tile directly into LDS ----
#pragma unroll
    for (int i = 0; i < 4; ++i) {
      const int q   = tid + i * 256;   // float4 id 0..1023
      const int row = q >> 3;          // 0..127
      const int c4  = q & 7;           // 0..7
      const unsigned loff =
          (unsigned)(uintptr_t)&lds_w[row * kStr + c4 * 4];  // 16B aligned
      async_wtile_b128(loff, W + row * kRdim + rbase + c4 * 4);
    }

    // ---- store im2col tile transposed: [p][r] ----
#pragma unroll
    for (int i = 0; i < 8; ++i) {
      const int rr = lr0 + i * 4;
      lds_xt[lp * kStr + rr] = xreg[i];
    }

    wait_asynccnt0();   // this wave's weight tiles have landed in LDS
    __syncthreads();    // everyone's weights + x stores visible

    // ---- compute: 8 K-steps of 4, 2x2 accumulator grid per wave ----
#pragma unroll
    for (int rb = 0; rb < kKC; rb += 4) {
      const int rl = rb + 2 * hf;  // A/B VGPR layout: vgpr v holds K = rl + v
      v2f a0 = *reinterpret_cast<const v2f*>(&lds_w[(m00 + idx) * kStr + rl]);
      v2f a1 = *reinterpret_cast<const v2f*>(&lds_w[(m00 + 16 + idx) * kStr + rl]);
      v2f b0 = *reinterpret_cast<const v2f*>(&lds_xt[(p00 + idx) * kStr + rl]);
      v2f b1 = *reinterpret_cast<const v2f*>(&lds_xt[(p00 + 16 + idx) * kStr + rl]);
      c00 = __builtin_amdgcn_wmma_f32_16x16x4_f32(false, a0, false, b0,
                                                  (short)0, c00, false, false);
      c01 = __builtin_amdgcn_wmma_f32_16x16x4_f32(false, a0, false, b1,
                                                  (short)0, c01, false, false);
      c10 = __builtin_amdgcn_wmma_f32_16x16x4_f32(false, a1, false, b0,
                                                  (short)0, c10, false, false);
      c11 = __builtin_amdgcn_wmma_f32_16x16x4_f32(false, a1, false, b1,
                                                  (short)0, c11, false, false);
    }
  }

  // ---- epilogue: add per-channel offset, scatter to NCHW ----
  const int n0  = pbase + p00 + idx;
  const int n1  = n0 + 16;
  const int bb0 = n0 / kSP;
  const int ss0 = n0 - bb0 * kSP;
  const int bb1 = n1 / kSP;
  const int ss1 = n1 - bb1 * kSP;

#pragma unroll
  for (int g = 0; g < 8; ++g) {
    const int mA = m00 + 8 * hf + g;       // C/D layout: vgpr g, halves M / M+8
    const int mB = mA + 16;
    const float oA = Off[mA];
    const float oB = Off[mB];
    Out[(bb0 * kK + mA) * kSP + ss0] = c00[g] + oA;
    Out[(bb1 * kK + mA) * kSP + ss1] = c01[g] + oA;
    Out[(bb0 * kK + mB) * kSP + ss0] = c10[g] + oB;
    Out[(bb1 * kK + mB) * kSP + ss1] = c11[g] + oB;
  }
}

extern "C" void kernel_launch(void* const* d_in, const int* in_sizes, int n_in,
                              void* d_out, int out_size, void* d_ws, size_t ws_size,
                              hipStream_t stream) {
  (void)in_sizes; (void)n_in; (void)out_size; (void)d_ws; (void)ws_size;
  const float* x   = (const float*)d_in[0];
  const float* w   = (const float*)d_in[1];
  const float* off = (const float*)d_in[2];
  float* out = (float*)d_out;

  dim3 grid(kNtot / kNP);  // 1568 blocks, exact
  dim3 block(256);         // 8 wave32 waves
  conv3x3_wmma_f32<<<grid, block, 0, stream>>>(x, w, off, out);
}